// ClusterControl_90348932038710
// MI455X (gfx1250) — compile-verified
//
#include <hip/hip_runtime.h>
#include <math.h>

// ---------------------------------------------------------------------------
// ClusterControl on MI455X (gfx1250, wave32).
//   d_in[0]: encodings  [8192,64] f32
//   d_in[1]: categorical[8192,25] f32
//   d_in[2]: k (=15, compile-time; k+1 = 16 tracked neighbors)
//   d_out (f32): enc[524288] | neigh_ent[8192] | cse[1] | npop[1] | max_groups[8192]
//
// Gram matrix via V_WMMA_F32_16X16X4_F32 (fp32 keeps the order statistic
// exact). Per wave: 16 rows, streaming top-16 selection (avoids the 256MB
// d2 matrix). B fragments are double-buffered in registers so loads overlap
// WMMA + selection; K is split into two independent WMMA chains for ILP;
// both lane-halves do selection on disjoint column halves, merged at the end.
// ---------------------------------------------------------------------------

typedef float v2f __attribute__((ext_vector_type(2)));
typedef float v8f __attribute__((ext_vector_type(8)));

#define B_N   8192
#define D_K   64
#define NCOMP 25
#define KSEL  16        // k+1 tracked smallest (k = 15)
#define EPS_E 1e-5f

__global__ void init_counts_kernel(unsigned* __restrict__ g) {
  if (threadIdx.x < NCOMP) g[threadIdx.x] = 0u;
}

__global__ void prep_kernel(const float* __restrict__ E,
                            const float* __restrict__ cat,
                            float* __restrict__ sq,
                            int* __restrict__ lab,
                            unsigned* __restrict__ gcounts,
                            float* __restrict__ out_max) {
  int i = blockIdx.x * blockDim.x + threadIdx.x;
  if (i >= B_N) return;
  const float* c = cat + (size_t)i * NCOMP;
  float mx = c[0];
  int mi = 0;
  #pragma unroll
  for (int j = 1; j < NCOMP; ++j) {     // strict '>' => first index wins ties
    float v = c[j];
    if (v > mx) { mx = v; mi = j; }
  }
  lab[i] = mi;
  out_max[i] = mx;
  atomicAdd(&gcounts[mi], 1u);

  const float* e = E + (size_t)i * D_K;
  float s = 0.f;
  #pragma unroll 8
  for (int j = 0; j < D_K; ++j) s = fmaf(e[j], e[j], s);
  sq[i] = s;
}

__global__ void copy_enc_kernel(const float4* __restrict__ src,
                                float4* __restrict__ dst) {
  int i = blockIdx.x * blockDim.x + threadIdx.x;  // 131072 float4 total
  dst[i] = src[i];
}

__global__ void global_entropy_kernel(const unsigned* __restrict__ gcounts,
                                      float* __restrict__ out2) {
  if (threadIdx.x == 0) {
    float ent = 0.f, npop = 0.f;
    for (int c = 0; c < NCOMP; ++c) {
      float gc = (float)gcounts[c];
      float b  = gc / (float)B_N;
      ent -= b * logf(b + EPS_E);
      if (gcounts[c] > 0u) npop += 1.f;
    }
    out2[0] = ent;   // cluster_size_entropy
    out2[1] = npop;  // n_populated
  }
}

// Load the 16 v2f B-fragments of one 16-col tile into registers (batched).
__device__ __forceinline__ void load_bfrags(const float* __restrict__ E,
                                            int colbase, int n16, int h,
                                            v2f* __restrict__ bf) {
  const float* br = E + (size_t)(colbase + n16) * D_K;
  #pragma unroll
  for (int c = 0; c < 16; ++c) {
    bf[c].x = br[4 * c + 2 * h];
    bf[c].y = br[4 * c + 2 * h + 1];
  }
}

// One wave per block; wave owns 16 rows.
__global__ __launch_bounds__(32) void knn_entropy_kernel(
    const float* __restrict__ E,
    const float* __restrict__ sq,
    const int*   __restrict__ lab,
    float*       __restrict__ out_ent) {
  const int lane    = threadIdx.x;    // 0..31
  const int h       = lane >> 4;      // half-select
  const int n16     = lane & 15;
  const int rowbase = blockIdx.x * 16;

  // A-tile fragments: lane holds E[rowbase+n16][4c + 2h + {0,1}] per chunk c.
  v2f afrag[16];
  {
    const float* ar = E + (size_t)(rowbase + n16) * D_K;
    #pragma unroll
    for (int c = 0; c < 16; ++c) {
      afrag[c].x = ar[4 * c + 2 * h];
      afrag[c].y = ar[4 * c + 2 * h + 1];
    }
  }

  // sq for the 8 rows this lane's C registers map to (C: VGPR v -> M = v+8h).
  float sqm[8];
  #pragma unroll
  for (int v = 0; v < 8; ++v) sqm[v] = sq[rowbase + v + 8 * h];

  // Per-lane sorted top-16: lane (h, n16) covers row n16, columns j in
  // [8h, 8h+8) of every tile. Lists merged across the two halves at the end.
  float best[KSEL];
  int   blab[KSEL];
  #pragma unroll
  for (int t = 0; t < KSEL; ++t) { best[t] = 3.4e38f; blab[t] = 0; }

  __shared__ float tile[16][17];      // d2 staging, +1 pad
  __shared__ float md[32][KSEL];      // end-of-loop merge buffers
  __shared__ int   ml[32][KSEL];

  // Double-buffered B fragments (constant-indexed via unrolled phase loop).
  v2f bf[2][16];
  load_bfrags(E, 0, n16, h, bf[0]);

  for (int nb = 0; nb < B_N; nb += 32) {
    #pragma unroll
    for (int ph = 0; ph < 2; ++ph) {
      const int nbc = nb + 16 * ph;
      int nbn = nbc + 16;
      if (nbn >= B_N) nbn = 0;        // harmless dummy prefetch on last tile

      // Two independent WMMA chains (even/odd K-chunks) for ILP.
      v8f acc0 = {0.f, 0.f, 0.f, 0.f, 0.f, 0.f, 0.f, 0.f};
      v8f acc1 = {0.f, 0.f, 0.f, 0.f, 0.f, 0.f, 0.f, 0.f};
      #pragma unroll
      for (int c = 0; c < 8; ++c) {
        acc0 = __builtin_amdgcn_wmma_f32_16x16x4_f32(
            false, afrag[2 * c],     false, bf[ph][2 * c],
            (short)0, acc0, false, false);
        acc1 = __builtin_amdgcn_wmma_f32_16x16x4_f32(
            false, afrag[2 * c + 1], false, bf[ph][2 * c + 1],
            (short)0, acc1, false, false);
      }

      // Prefetch next tile's fragments; latency overlaps d2/LDS/selection.
      load_bfrags(E, nbn, n16, h, bf[ph ^ 1]);

      // d2[M][N] = sq_M + sq_N - 2*dot, clamped at 0.
      float sqn = sq[nbc + n16];
      #pragma unroll
      for (int v = 0; v < 8; ++v) {
        float d2 = sqm[v] + sqn - 2.0f * (acc0[v] + acc1[v]);
        tile[v + 8 * h][n16] = fmaxf(d2, 0.0f);
      }
      __syncthreads();

      // All 32 lanes select: row n16, column half [8h, 8h+8).
      #pragma unroll
      for (int j = 0; j < 8; ++j) {
        const int col = 8 * h + j;
        float d = tile[n16][col];
        if (d < best[KSEL - 1]) {
          best[KSEL - 1] = d;
          blab[KSEL - 1] = lab[nbc + col];
          #pragma unroll
          for (int t = KSEL - 1; t > 0; --t) {   // bubble into sorted position
            if (best[t] < best[t - 1]) {
              float td = best[t]; best[t] = best[t - 1]; best[t - 1] = td;
              int   tl = blab[t]; blab[t] = blab[t - 1]; blab[t - 1] = tl;
            }
          }
        }
      }
      __syncthreads();
    }
  }

  // Merge the two half-column lists per row (lane r with lane r+16).
  #pragma unroll
  for (int t = 0; t < KSEL; ++t) { md[lane][t] = best[t]; ml[lane][t] = blab[t]; }
  __syncthreads();

  if (lane < 16) {
    #pragma unroll
    for (int t = 0; t < KSEL; ++t) {
      float d = md[lane + 16][t];
      int   g = ml[lane + 16][t];
      if (d < best[KSEL - 1]) {
        best[KSEL - 1] = d;
        blab[KSEL - 1] = g;
        #pragma unroll
        for (int u = KSEL - 1; u > 0; --u) {
          if (best[u] < best[u - 1]) {
            float td = best[u]; best[u] = best[u - 1]; best[u - 1] = td;
            int   tl = blab[u]; blab[u] = blab[u - 1]; blab[u - 1] = tl;
          }
        }
      }
    }

    // kth = (k+1)-th smallest; strictly-smaller entries are exactly the
    // tracked ones with best[t] < kth.
    float kth = best[KSEL - 1];
    float n = 0.f;
    #pragma unroll
    for (int t = 0; t < KSEL; ++t) n += (best[t] < kth) ? 1.f : 0.f;
    float inv = 1.0f / fmaxf(n, 1.0f);
    float ent = 0.f;
    #pragma unroll
    for (int cc = 0; cc < NCOMP; ++cc) {
      float cnt = 0.f;
      #pragma unroll
      for (int t = 0; t < KSEL; ++t)
        cnt += ((best[t] < kth) && (blab[t] == cc)) ? 1.f : 0.f;
      float b = cnt * inv;
      ent -= b * logf(b + EPS_E);   // zero bins contribute 0, matching ref
    }
    out_ent[rowbase + lane] = ent;
  }
}

extern "C" void kernel_launch(void* const* d_in, const int* in_sizes, int n_in,
                              void* d_out, int out_size, void* d_ws, size_t ws_size,
                              hipStream_t stream) {
  (void)in_sizes; (void)n_in; (void)out_size; (void)ws_size;
  const float* E   = (const float*)d_in[0];
  const float* cat = (const float*)d_in[1];
  // d_in[2] = k (=15) — baked in as KSEL-1 at compile time.

  float* out      = (float*)d_out;
  float* out_enc  = out;                       // [0, 524288)
  float* out_ne   = out + (size_t)B_N * D_K;   // [524288, 532480)
  float* out_cse  = out_ne + B_N;              // cse, npop
  float* out_mx   = out_cse + 2;               // [532482, 540674)

  float*    sq      = (float*)d_ws;
  int*      lab     = (int*)((char*)d_ws + (size_t)B_N * 4);
  unsigned* gcounts = (unsigned*)((char*)d_ws + (size_t)B_N * 8);

  init_counts_kernel<<<1, 32, 0, stream>>>(gcounts);
  prep_kernel<<<B_N / 256, 256, 0, stream>>>(E, cat, sq, lab, gcounts, out_mx);
  copy_enc_kernel<<<(B_N * D_K / 4) / 256, 256, 0, stream>>>(
      (const float4*)E, (float4*)out_enc);
  global_entropy_kernel<<<1, 32, 0, stream>>>(gcounts, out_cse);
  knn_entropy_kernel<<<B_N / 16, 32, 0, stream>>>(E, sq, lab, out_ne);
}